// GCNNet_8108898255422
// MI455X (gfx1250) — compile-verified
//
#include <hip/hip_runtime.h>
#include <math.h>

typedef __attribute__((ext_vector_type(2))) float v2f;
typedef __attribute__((ext_vector_type(8))) float v8f;

#define N_NODES 4096
#define NHEADS  4
#define DIN     512
#define DH1     256
#define DOUT    128
#define DPRED   64
#define NCLS    8
#define NTGT    512
#define MAXDEG  32

#define KCHUNK  64          // K-rows of B staged per LDS round
#define BSTRIDE 66          // padded LDS stride (even -> ds_load_b64 aligned)

// ---------------------------------------------------------------------------
// 1) Extract neighbor lists from the dense boolean mask (<=32 true per row).
// ---------------------------------------------------------------------------
__global__ void build_nbrs(const unsigned char* __restrict__ mask,
                           int* __restrict__ nbr, int* __restrict__ cnt) {
    int n = blockIdx.x * blockDim.x + threadIdx.x;
    if (n >= N_NODES) return;
    const unsigned char* row = mask + (size_t)n * N_NODES;
    int c = 0;
    for (int m = 0; m < N_NODES; ++m) {
        if (row[m] && c < MAXDEG) { nbr[n * MAXDEG + c] = m; ++c; }
    }
    cnt[n] = c;
}

// ---------------------------------------------------------------------------
// 2) Batched f32 GEMM with V_WMMA_F32_16X16X4_F32.
//    Block = 256 threads = 8 waves; each block computes a 128x16 strip of C.
//    The K x 16 B panel is staged (transposed) in LDS once per block and
//    shared by all 8 waves -> 8x less global B traffic, and B fragments are
//    constant-offset ds_load_b64 reads.  A fragments: contiguous
//    global_load_b64 off a bumped row pointer.  K must be a multiple of 64.
//
//    WMMA f32 16x16x4 layouts (ISA 7.12.2):
//      A-frag: lane%16 -> M, (lane/16)*2 -> K-half, 2 VGPRs/lane
//      B-frag: lane%16 -> N, (lane/16)*2 -> K-half, 2 VGPRs/lane
//      D:      VGPR r -> M = r + 8*(lane/16), N = lane%16
// ---------------------------------------------------------------------------
__global__ void gemm_wmma_f32(const float* __restrict__ A, long long strideAh, int lda,
                              const float* __restrict__ B, long long strideBh, int ldb,
                              const float* __restrict__ bias, int strideBiasH,
                              float* __restrict__ C, long long strideCh, int ldc,
                              int K) {
    __shared__ float Blds[16 * BSTRIDE];    // B panel, transposed: [n][k]

    const int tid  = threadIdx.x;
    const int lane = tid & 31;
    const int wave = tid >> 5;              // 0..7
    const int m0   = blockIdx.x * 128 + wave * 16;
    const int n0   = blockIdx.y * 16;
    const int h    = blockIdx.z;

    const float* Ah = A + (size_t)h * strideAh;
    const float* Bh = B + (size_t)h * strideBh;
    float*       Ch = C + (size_t)h * strideCh;

    const int mr = lane & 15;               // M within tile (A) / N within tile (B/D)
    const int kh = (lane >> 4) * 2;         // K-half select: 0 or 2

    const float* arow = Ah + (size_t)(m0 + mr) * lda + kh;  // bump by KCHUNK per round
    float*       bfrag = &Blds[mr * BSTRIDE + kh];          // bump target per round

    v8f acc = {0.f, 0.f, 0.f, 0.f, 0.f, 0.f, 0.f, 0.f};

    for (int k0 = 0; k0 < K; k0 += KCHUNK) {
        // prefetch the next A/B chunk into cache while we stage this one
        if (k0 + KCHUNK < K) {
            __builtin_prefetch(arow + KCHUNK, 0, 1);
            __builtin_prefetch(Bh + (size_t)(k0 + KCHUNK) * ldb + n0 + (tid & 15), 0, 1);
        }
        __syncthreads();                    // previous round's LDS reads done
        // stage B[k0..k0+63][n0..n0+15] transposed: Blds[n][k] (1024 elems / 256 thr)
        for (int e = tid; e < KCHUNK * 16; e += 256) {
            int r = e >> 4, c = e & 15;
            Blds[c * BSTRIDE + r] = Bh[(size_t)(k0 + r) * ldb + n0 + c];
        }
        __syncthreads();

#pragma unroll
        for (int kk = 0; kk < KCHUNK; kk += 4) {
            v2f a, b;
            a.x = arow[kk];                 // merges to global_load_b64
            a.y = arow[kk + 1];
            b.x = bfrag[kk];                // merges to ds_load_b64 (even offset)
            b.y = bfrag[kk + 1];
            acc = __builtin_amdgcn_wmma_f32_16x16x4_f32(
                /*neg_a=*/false, a, /*neg_b=*/false, b,
                /*c_mod=*/(short)0, acc, /*reuse_a=*/false, /*reuse_b=*/false);
        }
        arow += KCHUNK;
    }

    const int nc = lane & 15;
    const int mb = (lane >> 4) * 8;
    const float bv = bias ? bias[h * strideBiasH + n0 + nc] : 0.0f;
#pragma unroll
    for (int r = 0; r < 8; ++r) {
        Ch[(size_t)(m0 + mb + r) * ldc + n0 + nc] = acc[r] + bv;
    }
}

// ---------------------------------------------------------------------------
// 3) BatchNorm over the node axis.  Layout X: [G][Nrows][F]; one block/column.
// ---------------------------------------------------------------------------
__global__ void bn_stats(const float* __restrict__ X, int Nrows, int F,
                         float* __restrict__ mean, float* __restrict__ rstd) {
    const int col = blockIdx.x;             // col = g*F + o
    const int g = col / F, o = col % F;
    const float* base = X + (size_t)g * Nrows * F + o;
    float s = 0.f, sq = 0.f;
    for (int n = threadIdx.x; n < Nrows; n += blockDim.x) {
        float v = base[(size_t)n * F];
        s += v; sq += v * v;
    }
    __shared__ float ss[256], sqq[256];
    ss[threadIdx.x] = s; sqq[threadIdx.x] = sq;
    __syncthreads();
    for (int off = blockDim.x >> 1; off; off >>= 1) {
        if (threadIdx.x < off) {
            ss[threadIdx.x]  += ss[threadIdx.x + off];
            sqq[threadIdx.x] += sqq[threadIdx.x + off];
        }
        __syncthreads();
    }
    if (threadIdx.x == 0) {
        float m = ss[0] / (float)Nrows;
        float v = sqq[0] / (float)Nrows - m * m;   // biased variance (ddof=0)
        mean[col] = m;
        rstd[col] = rsqrtf(v + 1e-5f);
    }
}

__global__ void bn_apply(float* __restrict__ X,
                         const float* __restrict__ mean, const float* __restrict__ rstd,
                         const float* __restrict__ gamma, const float* __restrict__ beta,
                         int Nrows, int F, int total, int applyElu) {
    int i = blockIdx.x * blockDim.x + threadIdx.x;
    if (i >= total) return;
    int o = i % F;
    int g = i / (Nrows * F);
    int col = g * F + o;
    float v = (X[i] - mean[col]) * rstd[col] * gamma[col] + beta[col];
    if (applyElu) v = v > 0.f ? v : (expf(v) - 1.f);
    X[i] = v;
}

// ---------------------------------------------------------------------------
// 4) Sparse GAT attention: per (node, head) block. Scores only at the <=32
//    masked neighbors (softmax over neighbor set is exact), then weighted
//    row-gather aggregation.  F in {256, 128}.
// ---------------------------------------------------------------------------
__global__ void attn_sparse(const float* __restrict__ hbn,   // [H][N][F]
                            const float* __restrict__ Wa,    // [H][F][N]
                            const float* __restrict__ ba,    // [H][N]
                            const int* __restrict__ nbr, const int* __restrict__ cnt,
                            float* __restrict__ out,         // [H][N][F]
                            int F, int applyElu) {
    const int n = blockIdx.x;
    const int h = blockIdx.y;

    __shared__ float sh[DH1];       // current node's features (F <= 256)
    __shared__ float sc[MAXDEG];    // raw scores
    __shared__ float satt[MAXDEG];  // softmax weights
    __shared__ int   idx[MAXDEG];

    const float* hn = hbn + ((size_t)h * N_NODES + n) * F;
    for (int o = threadIdx.x; o < F; o += blockDim.x) sh[o] = hn[o];
    const int knum = cnt[n];
    if (threadIdx.x < knum) idx[threadIdx.x] = nbr[n * MAXDEG + threadIdx.x];
    __syncthreads();

    const int lane = threadIdx.x & 31;
    const int wave = threadIdx.x >> 5;      // 8 waves per block
    for (int j = wave; j < knum; j += 8) {
        const int m = idx[j];
        const float* wcol = Wa + (size_t)h * F * N_NODES + m;   // stride N per o
        float s = 0.f;
        for (int o = lane; o < F; o += 32) s += sh[o] * wcol[(size_t)o * N_NODES];
        for (int off = 16; off; off >>= 1) s += __shfl_down(s, off, 32);
        if (lane == 0) {
            s += ba[(size_t)h * N_NODES + m];
            s = (s >= 0.f) ? s : 0.2f * s;          // LeakyReLU(0.2)
            sc[j] = s;
        }
    }
    __syncthreads();

    if (threadIdx.x == 0) {                         // tiny softmax over <=32
        float mx = sc[0];
        for (int j = 1; j < knum; ++j) mx = fmaxf(mx, sc[j]);
        float sum = 0.f;
        for (int j = 0; j < knum; ++j) { float e = expf(sc[j] - mx); satt[j] = e; sum += e; }
        float inv = 1.f / sum;
        for (int j = 0; j < knum; ++j) satt[j] *= inv;
    }
    __syncthreads();

    float* op = out + ((size_t)h * N_NODES + n) * F;
    for (int o = threadIdx.x; o < F; o += blockDim.x) {
        float acc = 0.f;
        for (int j = 0; j < knum; ++j)
            acc += satt[j] * hbn[((size_t)h * N_NODES + idx[j]) * F + o];
        if (applyElu) acc = acc > 0.f ? acc : (expf(acc) - 1.f);
        op[o] = acc;
    }
}

// ---------------------------------------------------------------------------
// 5) Head average, gathered pred2 logits, NLL loss.
// ---------------------------------------------------------------------------
__global__ void head_avg(const float* __restrict__ X, float* __restrict__ Y, int NF) {
    int i = blockIdx.x * blockDim.x + threadIdx.x;
    if (i >= NF) return;
    float s = 0.f;
    for (int h = 0; h < NHEADS; ++h) s += X[(size_t)h * NF + i];
    Y[i] = s * (1.0f / NHEADS);
}

__global__ void pred2_logits(const float* __restrict__ y1,   // [N][PRED] (post BN+ELU)
                             const float* __restrict__ Wp2,  // [PRED][C]
                             const float* __restrict__ bp2,
                             const int* __restrict__ tX,
                             float* __restrict__ logits) {   // [T][C]
    int i = blockIdx.x * blockDim.x + threadIdx.x;
    if (i >= NTGT * NCLS) return;
    int t = i / NCLS, c = i % NCLS;
    const float* row = y1 + (size_t)tX[t] * DPRED;
    float s = bp2[c];
    for (int p = 0; p < DPRED; ++p) s += row[p] * Wp2[p * NCLS + c];
    logits[i] = s;
}

__global__ void nll_loss(const float* __restrict__ logits, const int* __restrict__ target,
                         float* __restrict__ lossOut) {
    __shared__ float red[NTGT];
    int t = threadIdx.x;
    const float* row = logits + (size_t)t * NCLS;
    float mx = row[0];
    for (int c = 1; c < NCLS; ++c) mx = fmaxf(mx, row[c]);
    float s = 0.f;
    for (int c = 0; c < NCLS; ++c) s += expf(row[c] - mx);
    red[t] = -(row[target[t]] - mx - logf(s));
    __syncthreads();
    for (int off = NTGT >> 1; off; off >>= 1) {
        if (t < off) red[t] += red[t + off];
        __syncthreads();
    }
    if (t == 0) lossOut[0] = red[0] / (float)NTGT;
}

// ---------------------------------------------------------------------------
// Orchestration.  Workspace layout (~34 MB): bufA 16MB | bufB 16MB | mean |
// rstd | nbr | cnt.
// ---------------------------------------------------------------------------
extern "C" void kernel_launch(void* const* d_in, const int* in_sizes, int n_in,
                              void* d_out, int out_size, void* d_ws, size_t ws_size,
                              hipStream_t stream) {
    const float* x     = (const float*)d_in[0];
    const unsigned char* mask = (const unsigned char*)d_in[1];
    const int*   tX    = (const int*)d_in[2];
    const int*   tgt   = (const int*)d_in[3];
    const float* W0    = (const float*)d_in[4];
    const float* b0    = (const float*)d_in[5];
    const float* g0    = (const float*)d_in[6];
    const float* beta0 = (const float*)d_in[7];
    const float* Wa0   = (const float*)d_in[8];
    const float* ba0   = (const float*)d_in[9];
    const float* W1    = (const float*)d_in[10];
    const float* b1    = (const float*)d_in[11];
    const float* g1    = (const float*)d_in[12];
    const float* beta1 = (const float*)d_in[13];
    const float* Wa1   = (const float*)d_in[14];
    const float* ba1   = (const float*)d_in[15];
    const float* bng   = (const float*)d_in[16];
    const float* bnb   = (const float*)d_in[17];
    const float* Wp1   = (const float*)d_in[18];
    const float* bp1   = (const float*)d_in[19];
    const float* gp1   = (const float*)d_in[20];
    const float* betap1= (const float*)d_in[21];
    const float* Wp2   = (const float*)d_in[22];
    const float* bp2   = (const float*)d_in[23];

    char* ws = (char*)d_ws;
    float* bufA = (float*)ws;                                   // 4M floats
    float* bufB = (float*)(ws + (size_t)16 * 1024 * 1024);      // 4M floats
    float* mean = (float*)(ws + (size_t)32 * 1024 * 1024);      // 1024 floats
    float* rstd = mean + 1024;
    int*   nbr  = (int*)(rstd + 1024);                          // N*32 ints
    int*   cnt  = nbr + N_NODES * MAXDEG;

    float* outp = (float*)d_out;          // [0] = loss, [1..] = logits [T][C]

    // neighbor extraction (reads mask once)
    build_nbrs<<<N_NODES / 256, 256, 0, stream>>>(mask, nbr, cnt);

    // ---- layer 0: h0 = x @ W0[h] + b0 -> bufA [H][N][256]
    gemm_wmma_f32<<<dim3(N_NODES / 128, DH1 / 16, NHEADS), 256, 0, stream>>>(
        x, 0LL, DIN, W0, (long long)DIN * DH1, DH1, b0, DH1,
        bufA, (long long)N_NODES * DH1, DH1, DIN);
    bn_stats<<<NHEADS * DH1, 256, 0, stream>>>(bufA, N_NODES, DH1, mean, rstd);
    bn_apply<<<(NHEADS * N_NODES * DH1) / 256, 256, 0, stream>>>(
        bufA, mean, rstd, g0, beta0, N_NODES, DH1, NHEADS * N_NODES * DH1, 0);
    // sparse attention + ELU -> bufB [H][N][256]
    attn_sparse<<<dim3(N_NODES, NHEADS), 256, 0, stream>>>(
        bufA, Wa0, ba0, nbr, cnt, bufB, DH1, 1);

    // ---- layer 1: h1 = out0 @ W1[h] + b1 -> bufA [H][N][128]
    gemm_wmma_f32<<<dim3(N_NODES / 128, DOUT / 16, NHEADS), 256, 0, stream>>>(
        bufB, (long long)N_NODES * DH1, DH1, W1, (long long)DH1 * DOUT, DOUT, b1, DOUT,
        bufA, (long long)N_NODES * DOUT, DOUT, DH1);
    bn_stats<<<NHEADS * DOUT, 256, 0, stream>>>(bufA, N_NODES, DOUT, mean, rstd);
    bn_apply<<<(NHEADS * N_NODES * DOUT) / 256, 256, 0, stream>>>(
        bufA, mean, rstd, g1, beta1, N_NODES, DOUT, NHEADS * N_NODES * DOUT, 0);
    // sparse attention, no activation -> bufB [H][N][128]
    attn_sparse<<<dim3(N_NODES, NHEADS), 256, 0, stream>>>(
        bufA, Wa1, ba1, nbr, cnt, bufB, DOUT, 0);

    // ---- head average -> bufA [N][128]; BN over nodes + ELU
    head_avg<<<(N_NODES * DOUT) / 256, 256, 0, stream>>>(bufB, bufA, N_NODES * DOUT);
    bn_stats<<<DOUT, 256, 0, stream>>>(bufA, N_NODES, DOUT, mean, rstd);
    bn_apply<<<(N_NODES * DOUT) / 256, 256, 0, stream>>>(
        bufA, mean, rstd, bng, bnb, N_NODES, DOUT, N_NODES * DOUT, 1);

    // ---- pred1: y1 = bufA @ Wp1 + bp1 -> bufB [N][64]; BN + ELU
    gemm_wmma_f32<<<dim3(N_NODES / 128, DPRED / 16, 1), 256, 0, stream>>>(
        bufA, 0LL, DOUT, Wp1, 0LL, DPRED, bp1, DPRED,
        bufB, 0LL, DPRED, DOUT);
    bn_stats<<<DPRED, 256, 0, stream>>>(bufB, N_NODES, DPRED, mean, rstd);
    bn_apply<<<(N_NODES * DPRED) / 256, 256, 0, stream>>>(
        bufB, mean, rstd, gp1, betap1, N_NODES, DPRED, N_NODES * DPRED, 1);

    // ---- pred2 only for the T gathered rows -> logits, then NLL loss
    pred2_logits<<<(NTGT * NCLS) / 256, 256, 0, stream>>>(bufB, Wp2, bp2, tX, outp + 1);
    nll_loss<<<1, NTGT, 0, stream>>>(outp + 1, tgt, outp);
}